// RelGraphConvLayer_78262894068349
// MI455X (gfx1250) — compile-verified
//
#include <hip/hip_runtime.h>
#include <hip/hip_bf16.h>

#define DFEAT 1024

typedef float v2f __attribute__((ext_vector_type(2)));
typedef float v8f __attribute__((ext_vector_type(8)));
typedef int   v4i __attribute__((ext_vector_type(4)));

#if __has_builtin(__builtin_amdgcn_global_load_async_to_lds_b128) && \
    __has_builtin(__builtin_amdgcn_s_wait_asynccnt)
#define USE_ASYNC_LDS 1
typedef __attribute__((address_space(1))) v4i* async_gptr;
typedef __attribute__((address_space(3))) v4i* async_lptr;
#else
#define USE_ASYNC_LDS 0
#endif

// ---------------------------------------------------------------- zero
__global__ void zero_kernel(float* __restrict__ p, long n) {
    long i = (long)blockIdx.x * blockDim.x + threadIdx.x;
    long stride = (long)gridDim.x * blockDim.x;
    for (; i < n; i += stride) p[i] = 0.0f;
}

// ---------------------------------------------------------------- scatter-add (segment_sum of gathered rows) + degree
// one block (256 threads) per edge; each thread handles one float4 of the 1024-float row
__global__ __launch_bounds__(256) void scatter_kernel(const float* __restrict__ xsrc,
                                                      const int* __restrict__ src,
                                                      const int* __restrict__ dst,
                                                      float* __restrict__ agg,
                                                      float* __restrict__ deg)
{
    const int e = blockIdx.x;
    const int s = src[e];
    const int d = dst[e];
    const float4 v = ((const float4*)(xsrc + (size_t)s * DFEAT))[threadIdx.x];
    float* ar = agg + (size_t)d * DFEAT + threadIdx.x * 4;
    atomicAdd(ar + 0, v.x);
    atomicAdd(ar + 1, v.y);
    atomicAdd(ar + 2, v.z);
    atomicAdd(ar + 3, v.w);
    if (threadIdx.x == 0) atomicAdd(deg + d, 1.0f);
}

// ---------------------------------------------------------------- normalize: agg[row,:] *= 1/max(deg,1)
__global__ __launch_bounds__(256) void normalize_kernel(float* __restrict__ agg,
                                                        const float* __restrict__ deg)
{
    const int row = blockIdx.x;
    const float s = 1.0f / fmaxf(deg[row], 1.0f);
    float4* p = (float4*)(agg + (size_t)row * DFEAT) + threadIdx.x;
    float4 v = *p;
    v.x *= s; v.y *= s; v.z *= s; v.w *= s;
    *p = v;
}

// ---------------------------------------------------------------- WMMA fp32 GEMM: out[row,col] (=|+=) A @ W + bias
// block: 256 threads = 8 waves. Block tile: 128 rows x 64 cols. Wave tile: 16 rows x 64 cols.
// W panels [128 x 64] double-buffered in LDS (2 x 32KB), staged with async
// global->LDS loads (ASYNCcnt) so panel k+1 streams in while panel k is consumed by WMMA.
__global__ __launch_bounds__(256) void gemm_bias_kernel(const float* __restrict__ A,
                                                        const float* __restrict__ W,
                                                        const float* __restrict__ bias,
                                                        float* __restrict__ out,
                                                        int nrows, int accumulate)
{
    __shared__ float ldsW[2][128 * 64];

    const int col0 = blockIdx.y * 64;
    const int lane = threadIdx.x & 31;
    const int wave = threadIdx.x >> 5;
    const int m    = lane & 15;          // row (A) / col (B) within 16
    const int kk   = (lane >> 4) << 1;   // lanes 0-15 -> K 0,1 ; lanes 16-31 -> K 2,3

    const int  row0    = (blockIdx.x * 8 + wave) * 16;
    const bool active  = (row0 + 16) <= nrows;
    const int  rowBase = active ? row0 : (nrows - 16);
    const float* arow  = A + (size_t)(rowBase + m) * DFEAT;

    v8f acc0 = {}, acc1 = {}, acc2 = {}, acc3 = {};

    // stage W[kc:kc+128, col0:col0+64] into LDS buffer `buf` (8 x b128 per thread)
    auto stage = [&](int buf, int kc) {
        #pragma unroll
        for (int it = 0; it < 8; ++it) {
            const int i  = it * 256 + threadIdx.x;  // float4 index
            const int r  = i >> 4;                  // 0..127
            const int c4 = (i & 15) << 2;           // 0..60
            const float* gp = W + (size_t)(kc + r) * DFEAT + col0 + c4;
            float* lp = &ldsW[buf][r * 64 + c4];
#if USE_ASYNC_LDS
            __builtin_amdgcn_global_load_async_to_lds_b128(
                (async_gptr)gp, (async_lptr)lp, 0, 0);
#else
            *(float4*)lp = *(const float4*)gp;
#endif
        }
    };

    stage(0, 0);

    for (int kc = 0; kc < DFEAT; kc += 128) {
        const int  buf  = (kc >> 7) & 1;
        const bool more = (kc + 128) < DFEAT;
        if (more) stage(buf ^ 1, kc + 128);
#if USE_ASYNC_LDS
        // async loads retire in order: <=8 outstanding means the current
        // panel's 8 ops are done; only the next panel's remain in flight.
        if (more) __builtin_amdgcn_s_wait_asynccnt(8);
        else      __builtin_amdgcn_s_wait_asynccnt(0);
#endif
        __syncthreads();

        const float* lw = &ldsW[buf][0];
        const float* ap = arow + kc;

        auto dostep = [&](v2f a, int k) {
            const float* l0 = lw + (k + kk) * 64;
            const float* l1 = l0 + 64;
            v2f b0; b0.x = l0[m];      b0.y = l1[m];
            v2f b1; b1.x = l0[16 + m]; b1.y = l1[16 + m];
            v2f b2; b2.x = l0[32 + m]; b2.y = l1[32 + m];
            v2f b3; b3.x = l0[48 + m]; b3.y = l1[48 + m];
            acc0 = __builtin_amdgcn_wmma_f32_16x16x4_f32(false, a, false, b0, (short)0, acc0, false, false);
            acc1 = __builtin_amdgcn_wmma_f32_16x16x4_f32(false, a, false, b1, (short)0, acc1, false, false);
            acc2 = __builtin_amdgcn_wmma_f32_16x16x4_f32(false, a, false, b2, (short)0, acc2, false, false);
            acc3 = __builtin_amdgcn_wmma_f32_16x16x4_f32(false, a, false, b3, (short)0, acc3, false, false);
        };

        // K window of 16 per group: 4 A-fragment loads in flight over 16 WMMAs
        #pragma unroll 2
        for (int k = 0; k < 128; k += 16) {
            const v2f a0 = *(const v2f*)(ap + k +  0 + kk);
            const v2f a1 = *(const v2f*)(ap + k +  4 + kk);
            const v2f a2 = *(const v2f*)(ap + k +  8 + kk);
            const v2f a3 = *(const v2f*)(ap + k + 12 + kk);
            dostep(a0, k + 0);
            dostep(a1, k + 4);
            dostep(a2, k + 8);
            dostep(a3, k + 12);
        }
        __syncthreads();
    }

    if (active) {
        // C/D layout: VGPR r -> lanes 0-15: (M=r, N=lane) ; lanes 16-31: (M=r+8, N=lane-16)
        const int n    = lane & 15;
        const int moff = (lane >> 4) * 8;
        const float bv0 = bias[col0 + n];
        const float bv1 = bias[col0 + 16 + n];
        const float bv2 = bias[col0 + 32 + n];
        const float bv3 = bias[col0 + 48 + n];
        #pragma unroll
        for (int r = 0; r < 8; ++r) {
            float* orow = out + (size_t)(row0 + moff + r) * DFEAT + col0;
            const float v0 = acc0[r] + bv0;
            const float v1 = acc1[r] + bv1;
            const float v2 = acc2[r] + bv2;
            const float v3 = acc3[r] + bv3;
            if (accumulate) {
                orow[n]      += v0;
                orow[16 + n] += v1;
                orow[32 + n] += v2;
                orow[48 + n] += v3;
            } else {
                orow[n]      = v0;
                orow[16 + n] = v1;
                orow[32 + n] = v2;
                orow[48 + n] = v3;
            }
        }
    }
}

// ---------------------------------------------------------------- host
extern "C" void kernel_launch(void* const* d_in, const int* in_sizes, int n_in,
                              void* d_out, int out_size, void* d_ws, size_t ws_size,
                              hipStream_t stream) {
    const float* x_drug = (const float*)d_in[0];
    const float* x_prot = (const float*)d_in[1];
    const float* W_ddi  = (const float*)d_in[2];
    const float* b_ddi  = (const float*)d_in[3];
    const float* W_dpi  = (const float*)d_in[4];
    const float* b_dpi  = (const float*)d_in[5];
    const float* W_ppi  = (const float*)d_in[6];
    const float* b_ppi  = (const float*)d_in[7];
    const int* src_ddi  = (const int*)d_in[8];
    const int* dst_ddi  = (const int*)d_in[9];
    const int* src_dpi  = (const int*)d_in[10];
    const int* dst_dpi  = (const int*)d_in[11];
    const int* src_ppi  = (const int*)d_in[12];
    const int* dst_ppi  = (const int*)d_in[13];

    const int n_drug = in_sizes[0] / DFEAT;   // 40000
    const int n_prot = in_sizes[1] / DFEAT;   // 40000
    const int n_edge = in_sizes[8];           // 320000

    float* out_drug = (float*)d_out;
    float* out_prot = out_drug + (size_t)n_drug * DFEAT;

    const int n_max = (n_drug > n_prot) ? n_drug : n_prot;
    float* agg = (float*)d_ws;                       // [n_max x 1024]
    float* deg = agg + (size_t)n_max * DFEAT;        // [n_max]

    auto run_rel = [&](const float* xsrc, const int* src, const int* dst,
                       int n_dst, const float* Wm, const float* bv,
                       float* outp, int accum) {
        zero_kernel<<<2048, 256, 0, stream>>>(agg, (long)n_dst * DFEAT);
        zero_kernel<<<64, 256, 0, stream>>>(deg, (long)n_dst);
        scatter_kernel<<<n_edge, 256, 0, stream>>>(xsrc, src, dst, agg, deg);
        normalize_kernel<<<n_dst, 256, 0, stream>>>(agg, deg);
        const int rowTiles = (n_dst + 15) / 16;
        dim3 ggrid((rowTiles + 7) / 8, DFEAT / 64);
        gemm_bias_kernel<<<ggrid, 256, 0, stream>>>(agg, Wm, bv, outp, n_dst, accum);
    };

    run_rel(x_drug, src_ddi, dst_ddi, n_drug, W_ddi, b_ddi, out_drug, 0);
    run_rel(x_drug, src_dpi, dst_dpi, n_prot, W_dpi, b_dpi, out_prot, 0);
    run_rel(x_prot, src_ppi, dst_ppi, n_prot, W_ppi, b_ppi, out_prot, 1);
}